// EdgeAwareGraphLayer_70231305224638
// MI455X (gfx1250) — compile-verified
//
#include <hip/hip_runtime.h>

typedef float v2f __attribute__((ext_vector_type(2)));
typedef float v8f __attribute__((ext_vector_type(8)));

#define N_NODES    100000
#define N_EDGES    1600000
#define NODE_DIM   128
#define EDGE_DIM   32
#define HIDDEN_DIM 128
#define MSG_K      (NODE_DIM + EDGE_DIM)    // 160
#define OUT_K      (NODE_DIM + HIDDEN_DIM)  // 256
#define MSG_STRIDE 164                       // 164 % 64 == 36 (=4*odd): conflict-free b64
#define OUT_STRIDE 260                       // 260 % 64 == 4  (=4*odd): conflict-free b64

// ---------------------------------------------------------------- zero agg
__global__ __launch_bounds__(256) void zero_f4_kernel(float4* __restrict__ p, int n4) {
    int i = blockIdx.x * blockDim.x + threadIdx.x;
    if (i < n4) p[i] = make_float4(0.f, 0.f, 0.f, 0.f);
}

// -------------------------------------------------- edge messages + scatter
// msgs = relu([src_feats | edge_feats] @ msg_w^T + msg_b);  agg[dst] += msgs
// 1 wave (32 lanes) computes 16 edges x 128 hidden via v_wmma_f32_16x16x4_f32.
__global__ __launch_bounds__(128) void edge_msg_kernel(
    const float* __restrict__ node_feats,   // [N,128]
    const int*   __restrict__ edge_src,     // [E]
    const int*   __restrict__ edge_dst,     // [E]
    const float* __restrict__ edge_feats,   // [E,32]
    const float* __restrict__ msg_w,        // [128,160] row-major
    const float* __restrict__ msg_b,        // [128]
    float*       __restrict__ agg)          // [N,128] (pre-zeroed)
{
    __shared__ __align__(16) float lds[4 * 16 * MSG_STRIDE];   // ~42 KB
    const int lane  = threadIdx.x & 31;
    const int wave  = threadIdx.x >> 5;
    const int ebase = (blockIdx.x * 4 + wave) * 16;            // exact: E = 100000*16
    float* tile = &lds[wave * 16 * MSG_STRIDE];

    // Stage A-tile: 16 edges x 160 (gather src node feats, then edge feats)
    for (int idx = lane; idx < 16 * MSG_K; idx += 32) {
        int e = idx / MSG_K;
        int k = idx - e * MSG_K;
        float v;
        if (k < NODE_DIM) {
            int s = edge_src[ebase + e];
            v = node_feats[s * NODE_DIM + k];
        } else {
            v = edge_feats[(ebase + e) * EDGE_DIM + (k - NODE_DIM)];
        }
        tile[e * MSG_STRIDE + k] = v;
    }
    asm volatile("s_wait_dscnt 0" ::: "memory");   // per-wave LDS visibility

    const int hi   = lane >> 4;        // 0: lanes 0-15, 1: lanes 16-31
    const int l16  = lane & 15;
    const int koff = hi * 2;           // A/B fragment K sub-offset per ISA layout

    v8f acc[8] = {};
    const float* arow = &tile[l16 * MSG_STRIDE + koff];

    for (int k = 0; k < MSG_K; k += 4) {
        v2f a = *(const v2f*)(arow + k);   // A: M=l16, K={k+koff, k+koff+1}
#pragma unroll
        for (int nt = 0; nt < 8; ++nt) {
            int n = nt * 16 + l16;         // B: N=n, K as above (B^T row of msg_w)
            v2f b = *(const v2f*)(msg_w + n * MSG_K + k + koff);
            acc[nt] = __builtin_amdgcn_wmma_f32_16x16x4_f32(
                false, a, false, b, (short)0, acc[nt], false, false);
        }
    }

    // Epilogue: bias + relu + scatter-add. D layout: VGPR i -> M = 8*hi + i, N = l16.
    const int mbase = hi * 8;
    int dsts[8];
#pragma unroll
    for (int i = 0; i < 8; ++i) dsts[i] = edge_dst[ebase + mbase + i];
#pragma unroll
    for (int nt = 0; nt < 8; ++nt) {
        int n = nt * 16 + l16;
        float bias = msg_b[n];
#pragma unroll
        for (int i = 0; i < 8; ++i) {
            float v = acc[nt][i] + bias;
            v = v > 0.f ? v : 0.f;
            unsafeAtomicAdd(&agg[dsts[i] * HIDDEN_DIM + n], v);  // global_atomic_add_f32 (L2-resident)
        }
    }
}

// ------------------------------------------------------------- node output
// out = relu([node_feats | agg] @ out_w^T + out_b), in-place over agg (=out).
__global__ __launch_bounds__(64) void node_out_kernel(
    const float* __restrict__ node_feats,   // [N,128]
    const float* __restrict__ out_w,        // [128,256] row-major
    const float* __restrict__ out_b,        // [128]
    float*       __restrict__ out)          // [N,128]: holds agg on entry
{
    __shared__ __align__(16) float lds[2 * 16 * OUT_STRIDE];   // ~33 KB
    const int lane  = threadIdx.x & 31;
    const int wave  = threadIdx.x >> 5;
    const int nbase = (blockIdx.x * 2 + wave) * 16;            // exact: N = 6250*16
    float* tile = &lds[wave * 16 * OUT_STRIDE];

    // Stage A-tile: 16 nodes x 256 ([node_feats | agg]); agg read before overwrite
    for (int idx = lane; idx < 16 * OUT_K; idx += 32) {
        int m = idx / OUT_K;
        int k = idx - m * OUT_K;
        float v;
        if (k < NODE_DIM) v = node_feats[(nbase + m) * NODE_DIM + k];
        else              v = out[(nbase + m) * HIDDEN_DIM + (k - NODE_DIM)];
        tile[m * OUT_STRIDE + k] = v;
    }
    asm volatile("s_wait_dscnt 0" ::: "memory");

    const int hi   = lane >> 4;
    const int l16  = lane & 15;
    const int koff = hi * 2;

    v8f acc[8] = {};
    const float* arow = &tile[l16 * OUT_STRIDE + koff];

    for (int k = 0; k < OUT_K; k += 4) {
        v2f a = *(const v2f*)(arow + k);
#pragma unroll
        for (int nt = 0; nt < 8; ++nt) {
            int n = nt * 16 + l16;
            v2f b = *(const v2f*)(out_w + n * OUT_K + k + koff);
            acc[nt] = __builtin_amdgcn_wmma_f32_16x16x4_f32(
                false, a, false, b, (short)0, acc[nt], false, false);
        }
    }

    const int mbase = hi * 8;
#pragma unroll
    for (int nt = 0; nt < 8; ++nt) {
        int n = nt * 16 + l16;
        float bias = out_b[n];
#pragma unroll
        for (int i = 0; i < 8; ++i) {
            float v = acc[nt][i] + bias;
            v = v > 0.f ? v : 0.f;
            out[(nbase + mbase + i) * NODE_DIM + n] = v;
        }
    }
}

// ------------------------------------------------------------------ launch
extern "C" void kernel_launch(void* const* d_in, const int* in_sizes, int n_in,
                              void* d_out, int out_size, void* d_ws, size_t ws_size,
                              hipStream_t stream) {
    const float* node_feats = (const float*)d_in[0];
    const int*   edge_index = (const int*)d_in[1];    // [2, E] int32
    const float* edge_feats = (const float*)d_in[2];
    const float* msg_w      = (const float*)d_in[3];
    const float* msg_b      = (const float*)d_in[4];
    const float* out_w      = (const float*)d_in[5];
    const float* out_b      = (const float*)d_in[6];
    float*       out        = (float*)d_out;          // doubles as agg accumulator

    const int* edge_src = edge_index;
    const int* edge_dst = edge_index + N_EDGES;

    // 1) agg (== d_out) = 0
    int n4 = (N_NODES * HIDDEN_DIM) / 4;               // 3.2M float4
    zero_f4_kernel<<<n4 / 256, 256, 0, stream>>>((float4*)out, n4);

    // 2) edge messages + scatter-add into agg
    edge_msg_kernel<<<N_EDGES / (4 * 16), 128, 0, stream>>>(
        node_feats, edge_src, edge_dst, edge_feats, msg_w, msg_b, out);

    // 3) node output GEMM, in place over agg
    node_out_kernel<<<N_NODES / (2 * 16), 64, 0, stream>>>(
        node_feats, out_w, out_b, out);
}